// TemporalPyramidPooling_8839042695721
// MI455X (gfx1250) — compile-verified
//
#include <hip/hip_runtime.h>

// Temporal pyramid pooling on MI455X (gfx1250, wave32).
// Sliding-window masked means formulated as a banded matmul on the f32 WMMA
// path: D[16 windows x 16 ch] += A[16x4 0/1 band] x B[4 t-rows x 16 ch].
// Each wave owns 16 windows x 64 channels (4 accumulators, 4 WMMAs/chunk),
// branch-free clamped loads, 1-deep software pipeline, compile-time W so the
// count tail fully unrolls into a load clause.

typedef float v2f __attribute__((ext_vector_type(2)));
typedef float v8f __attribute__((ext_vector_type(8)));

#define NACC 4  // 16-channel sub-tiles per wave -> 64 channels

template <int W>
__global__ __launch_bounds__(32)
void tpp_scale_wmma(const float* __restrict__ x,     // [B,T,D] f32
                    const int*   __restrict__ mask,  // [B,T]   0/1
                    float* __restrict__ avg_out,     // [B,S,D] f32
                    float* __restrict__ msk_out,     // [B,S]   f32 (valid>0)
                    int T, int D, int S)
{
    constexpr int ST  = W / 2;                       // stride
    constexpr int NCH = (15 * ST + W + 3) / 4;       // K-chunks per tile

    const int lane = threadIdx.x;            // one wave per block
    const int s0   = blockIdx.x * 16;        // window-tile start
    const int d0   = blockIdx.y * (16 * NACC);
    const int b    = blockIdx.z;

    const int mn     = lane & 15;            // M for A, N for B/C
    const int k0     = (lane >> 4) << 1;     // lanes<16 -> K=0,1 ; lanes>=16 -> K=2,3
    const int s_m    = s0 + mn;
    const int mvalid = (s_m < S) ? 1 : 0;
    const int wstart = ST * s_m;
    const int wend   = wstart + W;

    const int*   mb = mask + (long)b * T;
    const float* xb = x + (long)b * T * D + d0 + mn;

    v8f acc[NACC];
    #pragma unroll
    for (int j = 0; j < NACC; ++j) acc[j] = (v8f){};

    const int t_base = ST * s0;
    const int tmax   = T - 1;

    // ---- software pipeline prologue: load chunk 0 (clamped, branch-free) ----
    int t0  = t_base + k0;
    int tc0 = t0     < tmax ? t0     : tmax;
    int tc1 = t0 + 1 < tmax ? t0 + 1 : tmax;
    int   cm0 = mb[tc0];
    int   cm1 = mb[tc1];
    float cx0[NACC], cx1[NACC];
    #pragma unroll
    for (int j = 0; j < NACC; ++j) {
        cx0[j] = xb[tc0 * D + 16 * j];
        cx1[j] = xb[tc1 * D + 16 * j];
    }

    #pragma unroll 2
    for (int c = 0; c < NCH; ++c) {
        const int tcur = t_base + 4 * c + k0;

        // issue next chunk's loads early (addresses clamped -> always legal)
        const int tn  = tcur + 4;
        const int tn0 = tn     < tmax ? tn     : tmax;
        const int tn1 = tn + 1 < tmax ? tn + 1 : tmax;
        const int nm0 = mb[tn0];
        const int nm1 = mb[tn1];
        float nx0[NACC], nx1[NACC];
        #pragma unroll
        for (int j = 0; j < NACC; ++j) {
            nx0[j] = xb[tn0 * D + 16 * j];
            nx1[j] = xb[tn1 * D + 16 * j];
        }

        // A: 0/1 band weights for window s_m at times tcur, tcur+1
        const int in0 = mvalid & (tcur     >= wstart ? 1 : 0) & (tcur     < wend ? 1 : 0);
        const int in1 = mvalid & (tcur + 1 >= wstart ? 1 : 0) & (tcur + 1 < wend ? 1 : 0);
        const v2f A = { in0 ? 1.0f : 0.0f, in1 ? 1.0f : 0.0f };

        // B predicates: in-range time AND mask valid (pure selects, no branches)
        const int p0 = (tcur     < T ? 1 : 0) & (cm0 != 0 ? 1 : 0);
        const int p1 = (tcur + 1 < T ? 1 : 0) & (cm1 != 0 ? 1 : 0);

        #pragma unroll
        for (int j = 0; j < NACC; ++j) {
            const v2f Bv = { p0 ? cx0[j] : 0.0f, p1 ? cx1[j] : 0.0f };
            acc[j] = __builtin_amdgcn_wmma_f32_16x16x4_f32(
                         false, A, false, Bv, (short)0, acc[j], false, false);
        }

        cm0 = nm0; cm1 = nm1;
        #pragma unroll
        for (int j = 0; j < NACC; ++j) { cx0[j] = nx0[j]; cx1[j] = nx1[j]; }
    }

    // Per-window valid counts + reciprocal: lane L<16 owns window s0+L.
    // W is compile-time -> fully unrolled load clause, single wait.
    float inv = 0.0f;
    if (lane < 16 && mvalid) {
        const int* mw = mb + wstart;
        int cw[W];
        #pragma unroll
        for (int j = 0; j < W; ++j) cw[j] = mw[j];
        float cnt = 0.0f;
        #pragma unroll
        for (int j = 0; j < W; ++j) cnt += (cw[j] != 0) ? 1.0f : 0.0f;
        inv = 1.0f / fmaxf(cnt, 1e-6f);       // reference: num / clip(valid, EPS)
        if (d0 == 0)                           // bool output, once per (b, s)
            msk_out[(long)b * S + s_m] = (cnt > 0.0f) ? 1.0f : 0.0f;
    }

    // D layout: VGPR g -> row M = g + 8*(lane>=16), col N = lane&15
    const int rhalf = (lane >> 4) << 3;
    float* outb = avg_out + ((long)b * S) * D + d0 + mn;
    #pragma unroll
    for (int g = 0; g < 8; ++g) {
        const int   row  = g + rhalf;              // 0..15
        const float rinv = __shfl(inv, row, 32);   // broadcast 1/count for this row
        const int   s    = s0 + row;
        if (s < S) {
            #pragma unroll
            for (int j = 0; j < NACC; ++j)
                outb[(long)s * D + 16 * j] = acc[j][g] * rinv;
        }
    }
}

extern "C" void kernel_launch(void* const* d_in, const int* in_sizes, int n_in,
                              void* d_out, int out_size, void* d_ws, size_t ws_size,
                              hipStream_t stream) {
    (void)in_sizes; (void)n_in; (void)d_ws; (void)ws_size; (void)out_size;

    const float* x    = (const float*)d_in[0];   // [8,4096,512] f32
    const int*   mask = (const int*)d_in[1];     // [8,4096] bool -> int per harness rules
    float*       out  = (float*)d_out;

    const int B = 8, T = 4096, D = 512;

    const int S4  = (T - 4)  / 2 + 1;  // 2047
    const int S8  = (T - 8)  / 4 + 1;  // 1023
    const int S16 = (T - 16) / 8 + 1;  // 511

    // Flat output offsets in return order: avg4, avg8, avg16, msk4, msk8, msk16
    const long Oa4  = 0;
    const long Oa8  = Oa4  + (long)B * S4  * D;
    const long Oa16 = Oa8  + (long)B * S8  * D;
    const long Om4  = Oa16 + (long)B * S16 * D;
    const long Om8  = Om4  + (long)B * S4;
    const long Om16 = Om8  + (long)B * S8;

    dim3 blk(32, 1, 1);
    const int dTiles = D / (16 * NACC);      // 8
    dim3 g4 ((S4  + 15) / 16, dTiles, B);
    dim3 g8 ((S8  + 15) / 16, dTiles, B);
    dim3 g16((S16 + 15) / 16, dTiles, B);

    tpp_scale_wmma<4> <<<g4,  blk, 0, stream>>>(x, mask, out + Oa4,  out + Om4,  T, D, S4);
    tpp_scale_wmma<8> <<<g8,  blk, 0, stream>>>(x, mask, out + Oa8,  out + Om8,  T, D, S8);
    tpp_scale_wmma<16><<<g16, blk, 0, stream>>>(x, mask, out + Oa16, out + Om16, T, D, S16);
}